// HANLayer_26242250178589
// MI455X (gfx1250) — compile-verified
//
#include <hip/hip_runtime.h>
#include <hip/hip_bf16.h>

// ---------------- problem constants (match reference) ----------------
#define P_META   4
#define L_SUB    2
#define N_NODES  100000
#define N_EDGES  1600000
#define E_TABLE  200000
#define D        128          // D_IN == D_OUT
#define D_META   64
#define NREG     50000
#define SCALE_F  0.08838834764831845f   // 1/sqrt(128)

typedef __attribute__((ext_vector_type(2))) float v2f;
typedef __attribute__((ext_vector_type(8))) float v8f;

// ---------------- zero fill (float4 granularity) ----------------
__global__ void zero_f4(float4* __restrict__ p, int n4) {
    int i = blockIdx.x * blockDim.x + threadIdx.x;
    if (i < n4) p[i] = make_float4(0.f, 0.f, 0.f, 0.f);
}

// ---------------- gather x = E[eids] ----------------
// one wave per node, float4 per lane (512B per wave, coalesced)
__global__ void gather_rows(const float* __restrict__ E, const int* __restrict__ eids,
                            float* __restrict__ x) {
    int tid  = blockIdx.x * blockDim.x + threadIdx.x;
    int node = tid >> 5;
    int lane = tid & 31;
    if (node >= N_NODES) return;
    size_t r = (size_t)eids[node];
    float4 v = *(const float4*)(E + r * D + lane * 4);
    *(float4*)(x + (size_t)node * D + lane * 4) = v;
}

// ---------------- scatter: agg[dst] += x[src], cnt[dst] += 1 ----------------
// one wave per edge: 4 f32 atomics per lane; accumulator is L2-resident (51MB < 192MB)
__global__ void scatter_edges(const float* __restrict__ x,
                              const int* __restrict__ src, const int* __restrict__ dst,
                              float* __restrict__ agg, float* __restrict__ cnt) {
    int tid  = blockIdx.x * blockDim.x + threadIdx.x;
    int e    = tid >> 5;
    int lane = tid & 31;
    if (e >= N_EDGES) return;
    int s = src[e];
    int d = dst[e];
    float4 v = *(const float4*)(x + (size_t)s * D + lane * 4);
    float* a = agg + (size_t)d * D + lane * 4;
    atomicAdd(a + 0, v.x);
    atomicAdd(a + 1, v.y);
    atomicAdd(a + 2, v.z);
    atomicAdd(a + 3, v.w);
    if (lane == 0) atomicAdd(cnt + d, 1.0f);
}

// ---------------- fused normalize + dual GEMM + bias + relu ----------------
// h = (agg/max(cnt,1)) @ Wm + x @ Wr + bias, relu
// block = 256 threads = 8 waves; block owns 16 rows, wave w owns cols [16w,16w+16)
// V_WMMA_F32_16X16X4_F32, K swept 0..127 in steps of 4, two accumulation passes.
#define APITCH 132   // LDS row pitch in floats (16B-aligned, bank-spread fragment reads)

__global__ void rgcn_gemm_relu(const float* __restrict__ x,
                               const float* __restrict__ agg,
                               const float* __restrict__ cnt,
                               const float* __restrict__ Wr,   // [K=128][N=128] row major
                               const float* __restrict__ Wm,
                               const float* __restrict__ bias,
                               float* __restrict__ dstp, int dstStride) {
    __shared__ float sMsg[16 * APITCH];
    __shared__ float sRoot[16 * APITCH];

    const int t  = threadIdx.x;           // 0..255
    const int m0 = blockIdx.x * 16;

    // stage 16x128 strips of x and agg (normalized by cnt) into LDS
#pragma unroll
    for (int it = 0; it < 2; ++it) {
        int idx = t + it * 256;           // 0..511 : 16 rows x 32 float4 cols
        int row = idx >> 5;
        int c4  = idx & 31;
        size_t g = (size_t)(m0 + row) * D + c4 * 4;
        float4 vx = *(const float4*)(x + g);
        float4 va = *(const float4*)(agg + g);
        float inv = 1.0f / fmaxf(cnt[m0 + row], 1.0f);
        va.x *= inv; va.y *= inv; va.z *= inv; va.w *= inv;
        *(float4*)(sRoot + row * APITCH + c4 * 4) = vx;
        *(float4*)(sMsg  + row * APITCH + c4 * 4) = va;
    }
    __syncthreads();

    const int wave = t >> 5;
    const int lane = t & 31;
    const int n0   = wave * 16;
    const int rowA = lane & 15;
    const int koff = (lane >> 4) << 1;        // 0 for lanes 0-15, 2 for lanes 16-31
    const int colB = n0 + (lane & 15);

    v8f c = {};

#pragma unroll
    for (int pass = 0; pass < 2; ++pass) {
        const float* A = pass ? sRoot : sMsg;
        const float* B = pass ? Wr    : Wm;
#pragma unroll 8
        for (int kk = 0; kk < D; kk += 4) {
            // A fragment: 16x4, lane layout per ISA (VGPR0=K+0/K+2, VGPR1=K+1/K+3)
            v2f a = *(const v2f*)(A + rowA * APITCH + kk + koff);
            // B fragment: 4x16, mirrored K layout; weights are L2-hot (64KB each)
            v2f bfrag;
            bfrag.x = B[(size_t)(kk + koff)     * D + colB];
            bfrag.y = B[(size_t)(kk + koff + 1) * D + colB];
            c = __builtin_amdgcn_wmma_f32_16x16x4_f32(
                    false, a, false, bfrag, (short)0, c, false, false);
        }
    }

    // D layout: VGPR r -> (M = r [+8 for lanes>=16], N = lane&15)
    const float bv = bias[colB];
    const int rbase = m0 + ((lane >> 4) << 3);
#pragma unroll
    for (int r = 0; r < 8; ++r) {
        float v = fmaxf(c[r] + bv, 0.0f);
        dstp[(size_t)(rbase + r) * dstStride + colB] = v;
    }
}

// ---------------- query = metapath_emb @ Wq + bq  (tiny) ----------------
__global__ void compute_query(const float* __restrict__ me, const float* __restrict__ Wq,
                              const float* __restrict__ bq, float* __restrict__ query) {
    int d = threadIdx.x;   // 128 threads
#pragma unroll
    for (int p = 0; p < P_META; ++p) {
        float s = bq[d];
        for (int k = 0; k < D_META; ++k)
            s += me[p * D_META + k] * Wq[k * D + d];
        query[p * D + d] = s;
    }
}

// ---------------- softmax over metapaths + weighted combine ----------------
// one wave per node; lane handles dims {lane, lane+32, lane+64, lane+96}
__global__ void softmax_combine(const float* __restrict__ emb,   // [NREG][P][D]
                                const float* __restrict__ query, // [P][D]
                                float* __restrict__ out) {       // [NREG][D]
    int tid  = blockIdx.x * blockDim.x + threadIdx.x;
    int n    = tid >> 5;
    int lane = tid & 31;
    if (n >= NREG) return;
    const float* er = emb + (size_t)n * (P_META * D);

    float dots[P_META];
#pragma unroll
    for (int p = 0; p < P_META; ++p) {
        float s = 0.f;
#pragma unroll
        for (int cidx = 0; cidx < 4; ++cidx) {
            int d = lane + cidx * 32;
            s += query[p * D + d] * er[p * D + d];
        }
#pragma unroll
        for (int off = 16; off > 0; off >>= 1)
            s += __shfl_xor(s, off);
        dots[p] = s * SCALE_F;
    }
    float mx = dots[0];
#pragma unroll
    for (int p = 1; p < P_META; ++p) mx = fmaxf(mx, dots[p]);
    float w[P_META], sum = 0.f;
#pragma unroll
    for (int p = 0; p < P_META; ++p) { w[p] = __expf(dots[p] - mx); sum += w[p]; }
    float rs = 1.0f / sum;
#pragma unroll
    for (int p = 0; p < P_META; ++p) w[p] *= rs;

#pragma unroll
    for (int cidx = 0; cidx < 4; ++cidx) {
        int d = lane + cidx * 32;
        float o = 0.f;
#pragma unroll
        for (int p = 0; p < P_META; ++p) o += w[p] * er[p * D + d];
        out[(size_t)n * D + d] = o;
    }
}

// ---------------- launch ----------------
extern "C" void kernel_launch(void* const* d_in, const int* in_sizes, int n_in,
                              void* d_out, int out_size, void* d_ws, size_t ws_size,
                              hipStream_t stream) {
    const float* E    = (const float*)d_in[0];
    const float* me   = (const float*)d_in[1];
    const float* W_root = (const float*)d_in[2];   // [P][L][128][128]
    const float* W_rel  = (const float*)d_in[3];   // [P][L][128][128]
    const float* b      = (const float*)d_in[4];   // [P][L][128]
    const float* Wq     = (const float*)d_in[5];
    const float* bq     = (const float*)d_in[6];
    const int*   edge_index = (const int*)d_in[7]; // [P][2][N_EDGES]
    const int*   eids       = (const int*)d_in[8]; // [P][N_NODES]
    float* out = (float*)d_out;

    // workspace layout (floats)
    float* ws   = (float*)d_ws;
    const size_t NBUF = (size_t)N_NODES * D;       // 12.8M floats
    float* bufA  = ws;
    float* bufB  = ws + NBUF;
    float* cnt   = ws + 2 * NBUF;                  // N_NODES floats
    float* emb   = ws + 2 * NBUF + N_NODES;        // NREG*P*D floats
    float* query = emb + (size_t)NREG * P_META * D;

    const int ZB_BUF = (int)(NBUF / 4);            // float4 count
    const int ZB_CNT = N_NODES / 4;

    for (int i = 0; i < P_META; ++i) {
        const int* src = edge_index + ((size_t)i * 2 + 0) * N_EDGES;
        const int* dst = edge_index + ((size_t)i * 2 + 1) * N_EDGES;

        gather_rows<<<(N_NODES * 32) / 256, 256, 0, stream>>>(E, eids + (size_t)i * N_NODES, bufA);

        float* cur = bufA;
        float* nxt = bufB;
        for (int j = 0; j < L_SUB; ++j) {
            zero_f4<<<(ZB_BUF + 255) / 256, 256, 0, stream>>>((float4*)nxt, ZB_BUF);
            zero_f4<<<(ZB_CNT + 255) / 256, 256, 0, stream>>>((float4*)cnt, ZB_CNT);

            scatter_edges<<<(N_EDGES * 32) / 256, 256, 0, stream>>>(cur, src, dst, nxt, cnt);

            const float* Wr = W_root + ((size_t)(i * L_SUB + j)) * D * D;
            const float* Wm = W_rel  + ((size_t)(i * L_SUB + j)) * D * D;
            const float* bp = b      + ((size_t)(i * L_SUB + j)) * D;

            if (j < L_SUB - 1) {
                // write h in-place over agg (safe: strips staged to LDS first)
                rgcn_gemm_relu<<<N_NODES / 16, 256, 0, stream>>>(cur, nxt, cnt, Wr, Wm, bp,
                                                                 nxt, D);
                float* tmp = cur; cur = nxt; nxt = tmp;
            } else {
                // last sub-layer: only first NREG rows needed, strided into emb slab
                rgcn_gemm_relu<<<NREG / 16, 256, 0, stream>>>(cur, nxt, cnt, Wr, Wm, bp,
                                                              emb + (size_t)i * D, P_META * D);
            }
        }
    }

    compute_query<<<1, D, 0, stream>>>(me, Wq, bq, query);
    softmax_combine<<<(NREG * 32 + 255) / 256, 256, 0, stream>>>(emb, query, out);
}